// GlimpseModelFaster_33414845563392
// MI455X (gfx1250) — compile-verified
//
#include <hip/hip_runtime.h>
#include <hip/hip_bf16.h>

typedef __attribute__((ext_vector_type(2))) float v2f;
typedef __attribute__((ext_vector_type(8))) float v8f;

#define BATCH   32
#define IMGDIM  224
#define NK      144
#define KL      11
#define PAD     5
#define NPAIR   (BATCH * NK)     // 4608
#define NGROUP  (NPAIR / 16)     // 288 waves, 16 pairs each (16 | 144 -> one b per group)

// ---------------------------------------------------------------------------
// Kernel 1: per-kernel normalized 1-D Gaussian rows  wnorm[k][i], i=0..10
//   sigma_px = rint((sigma+1)/2*(H-1))   (round-half-even: 334.5 -> 334)
//   wnorm[i] = exp(-0.5*((i-5)/sigma_px)^2) / sum_i(...)
// k2d[k][i][j] == wnorm[i]*wnorm[j] is then already sum-normalized.
// ---------------------------------------------------------------------------
__global__ void glimpse_prep_weights(const float* __restrict__ sigma,
                                     float* __restrict__ wnorm) {
    int k = blockIdx.x * blockDim.x + threadIdx.x;
    if (k >= NK) return;
    float sp = rintf((sigma[k] + 1.0f) * 0.5f * (float)(IMGDIM - 1));
    float w[KL];
    float s = 0.0f;
#pragma unroll
    for (int i = 0; i < KL; ++i) {
        float n = (float)i - (float)PAD;
        float t = n / sp;
        w[i] = expf(-0.5f * t * t);
        s += w[i];
    }
    float inv = 1.0f / s;
#pragma unroll
    for (int i = 0; i < KL; ++i) wnorm[k * KL + i] = w[i] * inv;
}

// ---------------------------------------------------------------------------
// Kernel 2: one wave (32 lanes) per 16 (b,k) pairs.
// Batched 121-tap dot products via chained V_WMMA_F32_16X16X4_F32:
//   A row m = flattened 11x11 weights of pair m, B col m = gathered pixels,
//   D[m][m] = result. K padded 121 -> 124 (31 steps of 4).
// f32 A(16x4)/B(4x16) operand layout (ISA 7.12.2): lane L supplies M/N=L&15,
// K-in-step = 2*(L>>4)+{0,1}. K-slot permutation (free choice, consistent
// for A and B):  slot(t,half,u) <-> element  62*half + 31*u + t,
// so per lane the tap index is (62*half + compile-time-const): i,j reduce to
// one v_cndmask between constants instead of a dynamic-index select ladder.
// ---------------------------------------------------------------------------
__global__ void __launch_bounds__(32)
glimpse_wmma_kernel(const float* __restrict__ imgs,
                    const float* __restrict__ s_c,
                    const float* __restrict__ mu,
                    const float* __restrict__ wnorm,
                    float* __restrict__ out) {
    const int lane = threadIdx.x;            // 0..31, all active
    const bool hi  = (lane & 16) != 0;       // K-half selector
    const int m    = lane & 15;              // pair-in-group = matrix row/col
    const int p    = blockIdx.x * 16 + m;    // global pair id
    const int b    = p / NK;
    const int k    = p - b * NK;

    // Glimpse center in pixels (round-half-even like jnp.round).
    float cx = s_c[b * 2 + 0] + mu[k * 2 + 0];
    float cy = s_c[b * 2 + 1] + mu[k * 2 + 1];
    int x = (int)rintf((cx + 1.0f) * 0.5f * (float)(IMGDIM - 1));
    int y = (int)rintf((cy + 1.0f) * 0.5f * (float)(IMGDIM - 1));
    bool valid = (x >= 0) && (x < IMGDIM) && (y >= 0) && (y < IMGDIM);
    int x0 = min(max(x, 0), IMGDIM - 1) - PAD;   // top-left of window
    int y0 = min(max(y, 0), IMGDIM - 1) - PAD;

    // This pair's normalized 1-D Gaussian row (static-index register array).
    float wrow[KL];
#pragma unroll
    for (int i = 0; i < KL; ++i) wrow[i] = wnorm[k * KL + i];

    const float* imgb = imgs + (size_t)b * IMGDIM * IMGDIM;

    // ---- Phase 1: gather pixels + build weights, in consumption order ----
    // Lane handles elements e = 62*half + c for c = 0..61.
    // u=0 stream: c = t  (pixA/avA);  u=1 stream: c = 31+t (pixB/avB).
    float pixA[31], pixB[31], avA[31], avB[31];
#pragma unroll
    for (int t = 0; t < 31; ++t) {
#pragma unroll
        for (int u = 0; u < 2; ++u) {
            const int c   = 31 * u + t;        // compile-time
            const int eL  = c;                 // element if half==0 (<= 61)
            const int eH  = (62 + c > 120) ? 120 : (62 + c); // clamped index
            const bool padH = (62 + c > 120);  // K padding (half==1 only)
            const int iL = eL / KL, jL = eL % KL;   // all compile-time
            const int iH = eH / KL, jH = eH % KL;
            // weight: both variants are static-index; one select
            float aL = wrow[iL] * wrow[jL];
            float aH = padH ? 0.0f : wrow[iH] * wrow[jH];
            float a  = hi ? aH : aL;
            // tap coordinates: select between two inline constants
            int ii = hi ? iH : iL;
            int jj = hi ? jH : jL;
            int yy = y0 + ii;
            int xx = x0 + jj;
            bool inb = ((unsigned)yy < (unsigned)IMGDIM) &&
                       ((unsigned)xx < (unsigned)IMGDIM);
            int yyc = min(max(yy, 0), IMGDIM - 1);
            int xxc = min(max(xx, 0), IMGDIM - 1);
            float pv = imgb[yyc * IMGDIM + xxc];
            pv = inb ? pv : 0.0f;              // zero-padded "same" conv
            if (u == 0) { pixA[t] = pv; avA[t] = a; }
            else        { pixB[t] = pv; avB[t] = a; }
        }
    }

    // ---- Phase 2: 31 chained f32 WMMAs (K=4 each) ----
    v8f acc = {};
#pragma unroll
    for (int t = 0; t < 31; ++t) {
        v2f A;  A.x  = avA[t];  A.y  = avB[t];
        v2f Bv; Bv.x = pixA[t]; Bv.y = pixB[t];
        acc = __builtin_amdgcn_wmma_f32_16x16x4_f32(
            false, A, false, Bv, (short)0, acc, false, false);
    }

    // ---- Diagonal extraction per C/D layout: D[m][m] ----
    //   m in 0..7  -> lane m,    component m
    //   m in 8..15 -> lane m+16, component m-8
    bool writer = (lane < 8) || (lane >= 24);
    int  c      = (lane < 8) ? lane : (lane - 24);
    float r = acc[0];
#pragma unroll
    for (int q = 1; q < 8; ++q)
        if (c == q) r = acc[q];
    if (writer) out[p] = valid ? r : 0.0f;
}

// ---------------------------------------------------------------------------
extern "C" void kernel_launch(void* const* d_in, const int* in_sizes, int n_in,
                              void* d_out, int out_size, void* d_ws, size_t ws_size,
                              hipStream_t stream) {
    const float* imgs  = (const float*)d_in[0];  // (32,224,224)
    const float* s_c   = (const float*)d_in[1];  // (32,2)
    // d_in[2] = s_z, unused by forward
    const float* mu    = (const float*)d_in[3];  // (144,2)
    const float* sigma = (const float*)d_in[4];  // (144,)
    float* out   = (float*)d_out;                // (32,144)
    float* wnorm = (float*)d_ws;                 // 144*11 floats scratch

    glimpse_prep_weights<<<(NK + 63) / 64, 64, 0, stream>>>(sigma, wnorm);
    glimpse_wmma_kernel<<<NGROUP, 32, 0, stream>>>(imgs, s_c, mu, wnorm, out);
}